// Decoder_54623394071048
// MI455X (gfx1250) — compile-verified
//
#include <hip/hip_runtime.h>
#include <hip/hip_bf16.h>
#include <math.h>

// ---------------- problem dims ----------------
#define BB   16
#define SS   32
#define TKK  64
#define HH   256
#define EE   128
#define VV   50000
#define NEXX 100
#define NN   512          // 2H
#define MDLL 2048         // S*TK
#define VOUT (VV + NEXX)  // 50100

// ---------------- ws layout (float elements) ----------------
enum : int {
  WS_X      = 0,          // [16,128]   2048
  WS_GATES  = 2048,       // [16,1024]  16384
  WS_HNEW   = 18432,      // [16,256]   4096
  WS_CNEW   = 22528,      // [16,256]   4096
  WS_STHAT  = 26624,      // [16,512]   8192
  WS_DECF   = 34816,      // [16,512]   8192
  WS_SDECF  = 43008,      // [16,512]   8192
  WS_WATTN  = 51200,      // [512,64]   32768
  WS_SATTN  = 83968,      // [16,32]    512
  WS_OFFS   = 84480,      // [512] int  512
  WS_ATTNSW = 84992,      // [16,2048]  32768
  WS_CT     = 117760,     // [16,512]   8192
  WS_PGEN   = 125952,     // [16]       16
  WS_OUTH   = 125968,     // [16,256]   4096
  WS_END    = 130064
};

typedef __attribute__((ext_vector_type(2))) float v2f;
typedef __attribute__((ext_vector_type(8))) float v8f;

__device__ __forceinline__ float sigmoidf_(float x) { return 1.0f / (1.0f + __expf(-x)); }

// ------------------------------------------------------------------
// Wave-level fp32 WMMA: accumulate a 16x16 C tile of  A[16xK] * W^T
// where W is row-major [Nout x K]  (i.e. B[k][n] = W[n*ldw + k]).
// A-fragment layout (ISA 7.12.2, 32-bit A 16x4): lanes 0-15 hold K=0,1
// in v0,v1; lanes 16-31 hold K=2,3.  C/D: VGPR i = rows M=i / M=8+i.
// ------------------------------------------------------------------
__device__ __forceinline__ v8f wmma16_k(const float* __restrict__ A, int lda,
                                        const float* __restrict__ W, int ldw,
                                        int n0, int K, v8f acc) {
  const int lane = threadIdx.x & 31;
  const int r    = lane & 15;
  const int hi   = lane >> 4;
  const float* arow = A + r * lda;
  const float* wrow = W + (size_t)(n0 + r) * (size_t)ldw;
  for (int k0 = 0; k0 < K; k0 += 4) {
    v2f a, b;
    a.x = arow[k0 + 2 * hi];
    a.y = arow[k0 + 2 * hi + 1];
    const float2 bv = *(const float2*)(wrow + k0 + 2 * hi);
    b.x = bv.x; b.y = bv.y;
    acc = __builtin_amdgcn_wmma_f32_16x16x4_f32(false, a, false, b,
                                                (short)0, acc, false, false);
  }
  return acc;
}

__device__ __forceinline__ void wmma16_store(v8f acc, float* __restrict__ C,
                                             size_t ldc, int n0,
                                             const float* __restrict__ bias) {
  const int lane = threadIdx.x & 31;
  const int r    = lane & 15;
  const int hi   = lane >> 4;
  const float bb = bias ? bias[n0 + r] : 0.0f;
#pragma unroll
  for (int i = 0; i < 8; ++i)
    C[(size_t)(i + 8 * hi) * ldc + (size_t)(n0 + r)] = acc[i] + bb;
}

// ---------------- K0: zero attn_sw and c_t accumulators ----------------
__global__ void k_zero(float* ws) {
  int idx = blockIdx.x * 256 + threadIdx.x;
  if (idx < BB * MDLL)                ws[WS_ATTNSW + idx] = 0.0f;
  else if (idx < BB * MDLL + BB * NN) ws[WS_CT + idx - BB * MDLL] = 0.0f;
}

// ---------------- K1: embedding + x = [c_t_1, emb] @ W_xc^T + b ----------------
__global__ void k_embed_x(const int* __restrict__ y, const float* __restrict__ emb,
                          const float* __restrict__ ct1,
                          const float* __restrict__ W_xc, const float* __restrict__ b_xc,
                          float* __restrict__ xout) {
  __shared__ __align__(16) float A[16 * 640];
  const int tid = threadIdx.x;
  for (int i = tid; i < 16 * 640; i += 256) {
    int row = i / 640, col = i % 640;
    A[i] = (col < NN) ? ct1[row * NN + col]
                      : emb[(size_t)y[row] * EE + (col - NN)];
  }
  __syncthreads();
  const int wave = tid >> 5;              // 8 waves -> 8 tiles of 16 cols (E=128)
  v8f acc = {};
  acc = wmma16_k(A, 640, W_xc, 640, wave * 16, 640, acc);
  wmma16_store(acc, xout, EE, wave * 16, b_xc);
}

// ---------------- K2: LSTM gate GEMM: x@W_ih^T + h0@W_hh^T + biases ----------------
__global__ void k_gates(const float* __restrict__ x, const float* __restrict__ h0,
                        const float* __restrict__ W_ih, const float* __restrict__ W_hh,
                        const float* __restrict__ b_ih, const float* __restrict__ b_hh,
                        float* __restrict__ gates) {
  __shared__ __align__(16) float Ax[16 * 128];
  __shared__ __align__(16) float Ah[16 * 256];
  const int tid = threadIdx.x;
  for (int i = tid; i < 16 * 128; i += 256) Ax[i] = x[i];
  for (int i = tid; i < 16 * 256; i += 256) Ah[i] = h0[i];
  __syncthreads();
  const int wave = tid >> 5, lane = tid & 31, r = lane & 15, hi = lane >> 4;
  for (int t = wave; t < 64; t += 8) {     // 1024 cols / 16
    const int n0 = t * 16;
    v8f acc = {};
    acc = wmma16_k(Ax, 128, W_ih, 128, n0, 128, acc);
    acc = wmma16_k(Ah, 256, W_hh, 256, n0, 256, acc);
    const float bb = b_ih[n0 + r] + b_hh[n0 + r];
#pragma unroll
    for (int i = 0; i < 8; ++i)
      gates[(i + 8 * hi) * (4 * HH) + n0 + r] = acc[i] + bb;
  }
}

// ---------------- K3: LSTM activations -> h_new, c_new, s_t_hat ----------------
__global__ void k_lstm_act(const float* __restrict__ gates, const float* __restrict__ c0,
                           float* __restrict__ ws) {
  int idx = blockIdx.x * 256 + threadIdx.x;     // 16*256
  if (idx >= BB * HH) return;
  int b = idx >> 8, h = idx & 255;
  const float* g = gates + b * (4 * HH);
  float ig = sigmoidf_(g[h]);
  float fg = sigmoidf_(g[HH + h]);
  float gg = tanhf(g[2 * HH + h]);
  float og = sigmoidf_(g[3 * HH + h]);
  float c  = fg * c0[idx] + ig * gg;
  float hn = og * tanhf(c);
  ws[WS_HNEW + idx] = hn;
  ws[WS_CNEW + idx] = c;
  ws[WS_STHAT + b * NN + h]      = hn;
  ws[WS_STHAT + b * NN + HH + h] = c;
}

// ---------------- K4: dec_fea / sent_dec_fea GEMMs ----------------
__global__ void k_dec_features(const float* __restrict__ sthat,
                               const float* __restrict__ W_dec, const float* __restrict__ b_dec,
                               const float* __restrict__ W_sdec, const float* __restrict__ b_sdec,
                               float* __restrict__ decf, float* __restrict__ sdecf) {
  __shared__ __align__(16) float A[16 * 512];
  const int tid = threadIdx.x;
  for (int i = tid; i < 16 * 512; i += 256) A[i] = sthat[i];
  __syncthreads();
  const int wave = tid >> 5;
  for (int j = wave; j < 64; j += 8) {
    v8f acc = {};
    if (j < 32) {
      acc = wmma16_k(A, NN, W_dec, NN, j * 16, NN, acc);
      wmma16_store(acc, decf, NN, j * 16, b_dec);
    } else {
      int t = j - 32;
      acc = wmma16_k(A, NN, W_sdec, NN, t * 16, NN, acc);
      wmma16_store(acc, sdecf, NN, t * 16, b_sdec);
    }
  }
}

// ---------------- K5: word attention (one block per sentence) ----------------
// Streams 67MB of encoder_feature once with per-lane b128 loads.
__global__ void k_word_attn(const float* __restrict__ ef, const float* __restrict__ decf,
                            const float* __restrict__ vwg, const int* __restrict__ seq_lens2,
                            float* __restrict__ w_attn) {
  const int sent = blockIdx.x;
  const int doc  = sent >> 5;                    // // S
  const int tid  = threadIdx.x;
  __shared__ __align__(16) float df[NN];
  __shared__ __align__(16) float vw[NN];
  __shared__ float sc[TKK];
  __shared__ float smx, ssum;
  for (int i = tid; i < NN; i += 256) { df[i] = decf[doc * NN + i]; vw[i] = vwg[i]; }
  __syncthreads();
  const int wave = tid >> 5, lane = tid & 31;
  const float4* df4 = (const float4*)df;
  const float4* vw4 = (const float4*)vw;
  for (int w = wave; w < TKK; w += 8) {
    const float4* row4 = (const float4*)(ef + ((size_t)sent * TKK + w) * NN);
    float acc = 0.0f;
#pragma unroll
    for (int n4 = lane; n4 < NN / 4; n4 += 32) {       // 4 iters, b128 loads
      const float4 e = row4[n4];
      const float4 d = df4[n4];
      const float4 v = vw4[n4];
      acc += tanhf(e.x + d.x) * v.x + tanhf(e.y + d.y) * v.y
           + tanhf(e.z + d.z) * v.z + tanhf(e.w + d.w) * v.w;
    }
    for (int o = 16; o; o >>= 1) acc += __shfl_xor(acc, o, 32);
    if (lane == 0) sc[w] = acc;
  }
  __syncthreads();
  // softmax(s)*mask / sum == masked softmax (mask = k < len)
  if (tid == 0) {
    const int len = seq_lens2[sent];
    float mx = -1e30f;
    for (int k = 0; k < len; ++k) mx = fmaxf(mx, sc[k]);
    float sum = 0.0f;
    for (int k = 0; k < len; ++k) sum += __expf(sc[k] - mx);
    smx = mx; ssum = sum;
  }
  __syncthreads();
  if (tid < TKK) {
    const int len = seq_lens2[sent];
    w_attn[sent * TKK + tid] = (tid < len) ? __expf(sc[tid] - smx) / ssum : 0.0f;
  }
}

// ---------------- K6: sentence attention + exclusive-cumsum offsets ----------------
__global__ void k_sent_attn(const float* __restrict__ sef, const float* __restrict__ sdecf,
                            const float* __restrict__ svw, const float* __restrict__ mask,
                            const int* __restrict__ seq_lens2,
                            float* __restrict__ sattn, int* __restrict__ offs) {
  const int b = blockIdx.x, tid = threadIdx.x;
  __shared__ __align__(16) float df[NN];
  __shared__ __align__(16) float vw[NN];
  __shared__ float sc[SS];
  for (int i = tid; i < NN; i += 256) { df[i] = sdecf[b * NN + i]; vw[i] = svw[i]; }
  __syncthreads();
  const int wave = tid >> 5, lane = tid & 31;
  const float4* df4 = (const float4*)df;
  const float4* vw4 = (const float4*)vw;
  for (int s = wave; s < SS; s += 8) {
    const float4* row4 = (const float4*)(sef + (size_t)(b * SS + s) * NN);
    float acc = 0.0f;
#pragma unroll
    for (int n4 = lane; n4 < NN / 4; n4 += 32) {
      const float4 e = row4[n4];
      const float4 d = df4[n4];
      const float4 v = vw4[n4];
      acc += tanhf(e.x + d.x) * v.x + tanhf(e.y + d.y) * v.y
           + tanhf(e.z + d.z) * v.z + tanhf(e.w + d.w) * v.w;
    }
    for (int o = 16; o; o >>= 1) acc += __shfl_xor(acc, o, 32);
    if (lane == 0) sc[s] = acc;
  }
  __syncthreads();
  if (tid == 0) {
    float mx = -1e30f;
    for (int s = 0; s < SS; ++s) mx = fmaxf(mx, sc[s]);
    float e[SS]; float sum = 0.0f;
    for (int s = 0; s < SS; ++s) { e[s] = __expf(sc[s] - mx) * mask[b * SS + s]; sum += e[s]; }
    int run = 0;
    for (int s = 0; s < SS; ++s) {
      sattn[b * SS + s] = e[s] / sum;
      offs[b * SS + s]  = run;
      run += seq_lens2[b * SS + s];
    }
  }
}

// ---------------- K7: packed scatter of attn + c_t accumulation ----------------
// Streams 67MB of encoder_outputs once with per-thread b64 loads.
__global__ void k_combine(const float* __restrict__ enc_out, const float* __restrict__ w_attn,
                          const float* __restrict__ sattn, const int* __restrict__ offs,
                          const int* __restrict__ seq_lens2,
                          float* __restrict__ attn_sw, float* __restrict__ c_t) {
  const int sent = blockIdx.x;
  const int b    = sent >> 5;
  const int tid  = threadIdx.x;
  __shared__ float mult[TKK];
  const int   len = seq_lens2[sent];
  const int   off = offs[sent];
  const float sa  = sattn[sent];
  if (tid < TKK) {
    float m = (tid < len) ? sa * w_attn[sent * TKK + tid] : 0.0f;
    mult[tid] = m;
    if (tid < len) attn_sw[b * MDLL + off + tid] = m;   // packed placement (unique slots)
  }
  __syncthreads();
  const int n = tid * 2;                                 // float2 per thread, coalesced
  float a0 = 0.0f, a1 = 0.0f;
  const float* base = enc_out + (size_t)sent * TKK * NN + n;
  for (int k = 0; k < len; ++k) {
    const float  m = mult[k];
    const float2 e = *(const float2*)(base + (size_t)k * NN);
    a0 += m * e.x;
    a1 += m * e.y;
  }
  atomicAdd(&c_t[b * NN + n],     a0);
  atomicAdd(&c_t[b * NN + n + 1], a1);
}

// ---------------- K8: out1 GEMM + p_gen ----------------
__global__ void k_out1_pgen(const float* __restrict__ ws_f,
                            const float* __restrict__ W_out1, const float* __restrict__ b_out1,
                            const float* __restrict__ W_pgen, const float* __restrict__ b_pgen,
                            float* __restrict__ ws_mut) {
  __shared__ __align__(16) float A[16 * 768];
  const int tid = threadIdx.x;
  const float* hnew = ws_f + WS_HNEW;
  const float* ct   = ws_f + WS_CT;
  for (int i = tid; i < 16 * 768; i += 256) {
    int row = i / 768, col = i % 768;
    A[i] = (col < HH) ? hnew[row * HH + col] : ct[row * NN + (col - HH)];
  }
  __syncthreads();
  const int wave = tid >> 5;
  for (int t = wave; t < 16; t += 8) {      // 256 cols / 16
    v8f acc = {};
    acc = wmma16_k(A, 768, W_out1, 768, t * 16, 768, acc);
    wmma16_store(acc, ws_mut + WS_OUTH, HH, t * 16, b_out1);
  }
  __syncthreads();
  if (tid < BB) {
    const int b = tid;
    const float* sthat = ws_f + WS_STHAT;
    const float* xv    = ws_f + WS_X;
    float acc = b_pgen[0];
    for (int j = 0; j < NN; ++j) acc += ct[b * NN + j] * W_pgen[j];
    for (int j = 0; j < NN; ++j) acc += sthat[b * NN + j] * W_pgen[NN + j];
    for (int j = 0; j < EE; ++j) acc += xv[b * EE + j] * W_pgen[2 * NN + j];
    ws_mut[WS_PGEN + b] = sigmoidf_(acc);
  }
}

// ---------------- K9: vocab GEMM -> logits in d_out (stride VOUT) ----------------
__global__ void k_vocab_gemm(const float* __restrict__ outh,
                             const float* __restrict__ W_out2, const float* __restrict__ b_out2,
                             float* __restrict__ dout) {
  __shared__ __align__(16) float A[16 * 256];
  const int tid = threadIdx.x;
  for (int i = tid; i < 16 * 256; i += 256) A[i] = outh[i];
  __syncthreads();
  const int tile = blockIdx.x * 8 + (tid >> 5);
  if (tile >= VV / 16) return;              // 3125 tiles; wave-uniform guard
  const int n0 = tile * 16;
  {
    const int lane = tid & 31, r = lane & 15;
    __builtin_prefetch(&W_out2[(size_t)(n0 + r) * HH], 0, 1);   // stream W_out2 rows
  }
  v8f acc = {};
  acc = wmma16_k(A, HH, W_out2, HH, n0, HH, acc);
  wmma16_store(acc, dout, (size_t)VOUT, n0, b_out2);
}

// ---------------- K10: row softmax * p_gen + extended-vocab scatter ----------------
__global__ void k_softmax_final(float* __restrict__ dout, const float* __restrict__ pgen,
                                const float* __restrict__ attn_sw,
                                const int* __restrict__ ebev) {
  const int b = blockIdx.x, tid = threadIdx.x;
  float* row = dout + (size_t)b * VOUT;            // VOUT*4 bytes is 16B-aligned
  float4* row4 = (float4*)row;                     // 12500 float4s of logits
  __shared__ float red[256];
  float m = -1e30f;
  for (int v = tid; v < VV / 4; v += 256) {
    const float4 x = row4[v];
    m = fmaxf(m, fmaxf(fmaxf(x.x, x.y), fmaxf(x.z, x.w)));
  }
  red[tid] = m; __syncthreads();
  for (int s = 128; s; s >>= 1) { if (tid < s) red[tid] = fmaxf(red[tid], red[tid + s]); __syncthreads(); }
  const float rmax = red[0];
  __syncthreads();
  float sum = 0.0f;
  for (int v = tid; v < VV / 4; v += 256) {
    const float4 x = row4[v];
    sum += __expf(x.x - rmax) + __expf(x.y - rmax) + __expf(x.z - rmax) + __expf(x.w - rmax);
  }
  red[tid] = sum; __syncthreads();
  for (int s = 128; s; s >>= 1) { if (tid < s) red[tid] += red[tid + s]; __syncthreads(); }
  const float denom = red[0];
  const float pg = pgen[b];
  const float scale = pg / denom;
  for (int v = tid; v < VV / 4; v += 256) {
    float4 x = row4[v];
    x.x = __expf(x.x - rmax) * scale;
    x.y = __expf(x.y - rmax) * scale;
    x.z = __expf(x.z - rmax) * scale;
    x.w = __expf(x.w - rmax) * scale;
    row4[v] = x;
  }
  for (int v = VV + tid; v < VOUT; v += 256) row[v] = 0.0f;   // extra_zeros
  __threadfence();
  __syncthreads();
  const float om = 1.0f - pg;
  for (int j = tid; j < MDLL; j += 256) {
    const int idx = ebev[b * MDLL + j];
    atomicAdd(&row[idx], om * attn_sw[b * MDLL + j]);
  }
}

// ---------------- K11: emit aux outputs after final_dist ----------------
__global__ void k_emit_aux(const float* __restrict__ ws, const float* __restrict__ coverage,
                           float* __restrict__ dout) {
  const int idx = blockIdx.x * 256 + threadIdx.x;
  const int total = 4096 + 4096 + 8192 + 32768 + 16 + 32768;  // 81936
  if (idx >= total) return;
  float v;
  if      (idx < 4096)  v = ws[WS_HNEW + idx];
  else if (idx < 8192)  v = ws[WS_CNEW + idx - 4096];
  else if (idx < 16384) v = ws[WS_CT + idx - 8192];
  else if (idx < 49152) v = ws[WS_ATTNSW + idx - 16384];
  else if (idx < 49168) v = ws[WS_PGEN + idx - 49152];
  else                  v = coverage[idx - 49168];
  dout[(size_t)BB * VOUT + idx] = v;
}

// ------------------------------------------------------------------
extern "C" void kernel_launch(void* const* d_in, const int* in_sizes, int n_in,
                              void* d_out, int out_size, void* d_ws, size_t ws_size,
                              hipStream_t stream) {
  const int*   y_t_1   = (const int*)  d_in[0];
  const float* h0      = (const float*)d_in[1];
  const float* c0      = (const float*)d_in[2];
  const float* enc_out = (const float*)d_in[3];
  const float* enc_fea = (const float*)d_in[4];
  const int*   seqlen2 = (const int*)  d_in[5];
  // d_in[6] = sent_enc_outputs (unused by the reference)
  const float* sef     = (const float*)d_in[7];
  const float* smask   = (const float*)d_in[8];
  const float* coverage= (const float*)d_in[9];
  const float* c_t_1   = (const float*)d_in[10];
  // d_in[11] = extra_zeros (always zero; emitted directly)
  const int*   ebev    = (const int*)  d_in[12];
  const float* emb     = (const float*)d_in[13];
  const float* W_xc    = (const float*)d_in[14];
  const float* b_xc    = (const float*)d_in[15];
  const float* W_ih    = (const float*)d_in[16];
  const float* W_hh    = (const float*)d_in[17];
  const float* b_ih    = (const float*)d_in[18];
  const float* b_hh    = (const float*)d_in[19];
  const float* W_dec   = (const float*)d_in[20];
  const float* b_dec   = (const float*)d_in[21];
  const float* v_w     = (const float*)d_in[22];
  const float* W_sdec  = (const float*)d_in[23];
  const float* b_sdec  = (const float*)d_in[24];
  const float* sv_w    = (const float*)d_in[25];
  const float* W_pgen  = (const float*)d_in[26];
  const float* b_pgen  = (const float*)d_in[27];
  const float* W_out1  = (const float*)d_in[28];
  const float* b_out1  = (const float*)d_in[29];
  const float* W_out2  = (const float*)d_in[30];
  const float* b_out2  = (const float*)d_in[31];

  float* ws   = (float*)d_ws;
  int*   offs = (int*)(ws + WS_OFFS);
  float* dout = (float*)d_out;

  k_zero<<<(BB * MDLL + BB * NN + 255) / 256, 256, 0, stream>>>(ws);
  k_embed_x<<<1, 256, 0, stream>>>(y_t_1, emb, c_t_1, W_xc, b_xc, ws + WS_X);
  k_gates<<<1, 256, 0, stream>>>(ws + WS_X, h0, W_ih, W_hh, b_ih, b_hh, ws + WS_GATES);
  k_lstm_act<<<(BB * HH + 255) / 256, 256, 0, stream>>>(ws + WS_GATES, c0, ws);
  k_dec_features<<<1, 256, 0, stream>>>(ws + WS_STHAT, W_dec, b_dec, W_sdec, b_sdec,
                                        ws + WS_DECF, ws + WS_SDECF);
  k_word_attn<<<BB * SS, 256, 0, stream>>>(enc_fea, ws + WS_DECF, v_w, seqlen2,
                                           ws + WS_WATTN);
  k_sent_attn<<<BB, 256, 0, stream>>>(sef, ws + WS_SDECF, sv_w, smask, seqlen2,
                                      ws + WS_SATTN, offs);
  k_combine<<<BB * SS, 256, 0, stream>>>(enc_out, ws + WS_WATTN, ws + WS_SATTN, offs,
                                         seqlen2, ws + WS_ATTNSW, ws + WS_CT);
  k_out1_pgen<<<1, 256, 0, stream>>>(ws, W_out1, b_out1, W_pgen, b_pgen, ws);
  k_vocab_gemm<<<(VV / 16 + 7) / 8, 256, 0, stream>>>(ws + WS_OUTH, W_out2, b_out2, dout);
  k_softmax_final<<<BB, 256, 0, stream>>>(dout, ws + WS_PGEN, ws + WS_ATTNSW, ebev);
  k_emit_aux<<<(81936 + 255) / 256, 256, 0, stream>>>(ws, coverage, dout);
}